// spatial_72859825209401
// MI455X (gfx1250) — compile-verified
//
#include <hip/hip_runtime.h>

// Problem constants from the reference (filt_size is compile-time 8).
#define BB   16
#define CC   64
#define HH   256
#define WW   256
#define FF   8

// True 128-bit vector type (accepted by nontemporal builtins, unlike HIP float4).
typedef float v4f __attribute__((ext_vector_type(4)));

// One workgroup per (b, hb, wb-group-of-4): tile = 64ch x 8rows x 32floats = 16384 floats.
//  - x tile staged into LDS with CDNA5 async global->LDS B128 ops (ASYNCcnt);
//    32 lanes x 16B = 4 full 128B lines per wave instruction (perfectly coalesced)
//  - each thread consumes only its own staged chunks => no barrier needed after the
//    wave-local s_wait_asynccnt; the only workgroup barrier guards the red[] exchange
//  - mask streamed once with nontemporal loads, fused multiply into 4 per-block sums
//  - segmented wave32 reduction (xor 1,8,16) + 8-wave LDS combine per sub-block
//  - gated x replayed from LDS, nontemporal coalesced stores; x hits HBM exactly once.
__global__ __launch_bounds__(256) void spatial_gate_fused_kernel(
    const float* __restrict__ x,
    const float* __restrict__ mask,
    float* __restrict__ out)
{
    const int wbg = blockIdx.x;     // 0..7   (group of 4 wb blocks)
    const int hb  = blockIdx.y;     // 0..31
    const int b   = blockIdx.z;     // 0..15
    const int tid = threadIdx.x;    // 0..255

    __shared__ float xs[CC * FF * 32];  // 16384 floats = 64 KB, linear in chunkId
    __shared__ float red[8 * 4];        // [wave][sub-block]

    // chunkId(it) = it*256 + tid  =>  row = it*32 + (tid>>3), k = tid&7 (constant).
    // Per-thread base offset (floats); per-iteration stride = 32*WW floats (32 KB).
    const size_t base = ((size_t)b * CC * HH + (size_t)hb * FF) * WW + (size_t)wbg * 32;
    const size_t toff = base + (size_t)(tid >> 3) * WW + (size_t)(tid & 7) * 4;
    const unsigned ldsb = (unsigned)(uintptr_t)&xs[tid * 4];   // + it*4096 bytes

    // ---- stage x tile into LDS asynchronously (16 x B128 per thread) ----
    #pragma unroll
    for (int it = 0; it < 16; ++it) {
        const unsigned lds0 = ldsb + (unsigned)it * 4096u;
        const uint64_t g0   = (uint64_t)(uintptr_t)(x + toff + (size_t)it * (32 * WW));
        asm volatile("global_load_async_to_lds_b128 %0, %1, off"
                     :: "v"(lds0), "v"(g0) : "memory");
    }

    // wave-local drain: every chunk this thread reads below was staged by itself.
    asm volatile("s_wait_asynccnt 0x0" ::: "memory");

    // ---- fused reduce: stream mask once, multiply with LDS x ----
    float partial = 0.0f;
    #pragma unroll
    for (int it = 0; it < 16; ++it) {
        const v4f mq = __builtin_nontemporal_load(
                           (const v4f*)(mask + toff + (size_t)it * (32 * WW)));
        const v4f xq = *(const v4f*)&xs[tid * 4 + it * 1024];
        partial += xq[0] * mq[0] + xq[1] * mq[1] + xq[2] * mq[2] + xq[3] * mq[3];
    }

    // every chunk of this thread lies in sub-block blk = (tid&7)>>1.
    // reduce over the 8 lanes of a wave sharing that sub-block: xor masks 1, 8, 16.
    partial += __shfl_xor(partial, 1, 32);
    partial += __shfl_xor(partial, 8, 32);
    partial += __shfl_xor(partial, 16, 32);
    const int lane = tid & 31;
    const int wave = tid >> 5;
    if (lane < 8 && (lane & 1) == 0)
        red[wave * 4 + (lane >> 1)] = partial;
    __syncthreads();   // only cross-wave exchange in the kernel

    // final per-sub-block sum across the 8 waves; gate = (s > 0)
    const int blk = (tid & 7) >> 1;     // 0..3
    float s = 0.0f;
    #pragma unroll
    for (int w = 0; w < 8; ++w)
        s += red[w * 4 + blk];
    const float gate = (s > 0.0f) ? 1.0f : 0.0f;

    // ---- gated writeback: replay x from LDS, streaming coalesced stores ----
    #pragma unroll
    for (int it = 0; it < 16; ++it) {
        const v4f xq = *(const v4f*)&xs[tid * 4 + it * 1024];
        const v4f o  = xq * gate;
        __builtin_nontemporal_store(o, (v4f*)(out + toff + (size_t)it * (32 * WW)));
    }
}

extern "C" void kernel_launch(void* const* d_in, const int* in_sizes, int n_in,
                              void* d_out, int out_size, void* d_ws, size_t ws_size,
                              hipStream_t stream) {
    (void)in_sizes; (void)n_in; (void)d_ws; (void)ws_size; (void)out_size;
    const float* x    = (const float*)d_in[0];
    const float* mask = (const float*)d_in[1];
    // d_in[2] is filt_size (== 8), baked in at compile time.
    float* out = (float*)d_out;

    dim3 grid(WW / (FF * 4), HH / FF, BB);   // (8, 32, 16) = 4096 workgroups
    dim3 block(256, 1, 1);                   // 8 wave32s
    spatial_gate_fused_kernel<<<grid, block, 0, stream>>>(x, mask, out);
}